// GraphEncoder_12953621365355
// MI455X (gfx1250) — compile-verified
//
#include <hip/hip_runtime.h>
#include <hip/hip_bf16.h>
#include <math.h>

// ---------------------------------------------------------------------------
// GraphEncoder on MI455X (gfx1250): fully-connected Bayesian GCN ==
//   8 x [ (h @ W_k) 1024x64x64  then  (adj^T, diag=0) @ T 1024x64x1024 ]
// fp32 via V_WMMA_F32_16X16X4_F32 (wave32), staged with
// GLOBAL_LOAD_ASYNC_TO_LDS_B128 + double-buffered LDS (ASYNCcnt pipe).
// ---------------------------------------------------------------------------

typedef __attribute__((ext_vector_type(2))) float v2f;
typedef __attribute__((ext_vector_type(8))) float v8f;

#define NN   1024
#define FF   64
#define WSET 8
#define WELT (WSET * FF * FF)   // 32768

// ---------------------------------------------------------------------------
// 1) Reparameterized weights + KL partial sums (deterministic two-stage).
// ---------------------------------------------------------------------------
__global__ __launch_bounds__(256)
void prep_kernel(const float* __restrict__ Wmu, const float* __restrict__ Wrho,
                 const float* __restrict__ eps, float* __restrict__ W,
                 float* __restrict__ klpart) {
  int idx = blockIdx.x * 256 + threadIdx.x;     // exactly 32768 threads
  float mu  = Wmu[idx];
  float rho = Wrho[idx];
  float e   = eps[idx];
  float sp  = (rho > 20.0f) ? rho : log1pf(expf(rho));   // softplus
  W[idx] = mu + sp * e;
  // KL(N(mu,sp) || N(0,0.1)) elementwise: log(0.1/sp) + (sp^2+mu^2)/(2*0.01) - 0.5
  float kl = logf(0.1f / sp) + (sp * sp + mu * mu) * 50.0f - 0.5f;

  // wave32 reduce
  #pragma unroll
  for (int o = 16; o; o >>= 1) kl += __shfl_down(kl, o, 32);
  __shared__ float red[8];
  int lane = threadIdx.x & 31, wv = threadIdx.x >> 5;
  if (lane == 0) red[wv] = kl;
  __syncthreads();
  if (threadIdx.x == 0) {
    float s = 0.0f;
    #pragma unroll
    for (int w = 0; w < 8; ++w) s += red[w];
    klpart[blockIdx.x] = s;
  }
}

__global__ void kl_final_kernel(const float* __restrict__ klpart, float* __restrict__ klout) {
  if (threadIdx.x == 0 && blockIdx.x == 0) {
    float s = 0.0f;
    for (int i = 0; i < 128; ++i) s += klpart[i];   // fixed order: deterministic
    *klout = s;
  }
}

// ---------------------------------------------------------------------------
// 2) Bm[j,i] = adj[i,j] (diag=0), B2 = Bm^2  — LDS tiled transpose.
// ---------------------------------------------------------------------------
__global__ __launch_bounds__(256)
void transpose_kernel(const float* __restrict__ adj, float* __restrict__ Bm,
                      float* __restrict__ B2) {
  __shared__ float tile[32][33];
  int bx = blockIdx.x, by = blockIdx.y;
  int tx = threadIdx.x, ty = threadIdx.y;       // 32 x 8
  #pragma unroll
  for (int k = 0; k < 32; k += 8)
    tile[ty + k][tx] = adj[(by * 32 + ty + k) * NN + bx * 32 + tx];
  __syncthreads();
  #pragma unroll
  for (int k = 0; k < 32; k += 8) {
    int j = bx * 32 + ty + k;                   // output row  (= dst node)
    int i = by * 32 + tx;                       // output col  (= src node)
    float v = (i == j) ? 0.0f : tile[tx][ty + k];
    Bm[j * NN + i] = v;
    B2[j * NN + i] = v * v;
  }
}

// ---------------------------------------------------------------------------
// 3) WMMA f32 GEMM: out[1024x64] = A[1024xK] @ B[Kx64]   (ldb == 64)
//    4 waves / block, each wave owns one 16x16 tile; block = 16 rows of M.
//    K staged in 64-chunks via async global->LDS, double buffered.
//    mode 0: store raw.  mode 1: var=exp(v)+1e-6 -> out ; sqrt(var) -> out2.
// ---------------------------------------------------------------------------
#define APAD 68   // row pitch (floats): 272B rows -> 16B-aligned, conflict-free

__global__ __launch_bounds__(128)
void gemm_wmma_kernel(const float* __restrict__ A, int lda,
                      const float* __restrict__ B,
                      float* __restrict__ out, float* __restrict__ out2,
                      int K, int mode) {
  __shared__ __align__(16) float Ast[2][16 * APAD];
  __shared__ __align__(16) float Bst[2][64 * APAD];

  const int tid  = threadIdx.x;
  const int lane = tid & 31;
  const int wv   = tid >> 5;        // 0..3  -> N tile
  const int hl   = lane >> 4;       // half of wave
  const int lr   = lane & 15;
  const int m0   = blockIdx.x * 16;
  const int n0   = wv * 16;

  // Issue one K-chunk (64-wide) of async global->LDS b128 copies.
  auto stage = [&](int buf, int kb) {
    #pragma unroll
    for (int p = 0; p < 2; ++p) {               // A tile: 16 x 64
      int i4 = tid + p * 128;                   // 0..255
      int r = i4 >> 4, cc = (i4 & 15) * 4;
      unsigned l = (unsigned)(uintptr_t)&Ast[buf][r * APAD + cc];
      const float* g = &A[(m0 + r) * lda + kb + cc];
      asm volatile("global_load_async_to_lds_b128 %0, %1, off"
                   :: "v"(l), "v"(g) : "memory");
    }
    #pragma unroll
    for (int p = 0; p < 8; ++p) {               // B tile: 64 x 64
      int i4 = tid + p * 128;                   // 0..1023
      int r = i4 >> 4, cc = (i4 & 15) * 4;
      unsigned l = (unsigned)(uintptr_t)&Bst[buf][r * APAD + cc];
      const float* g = &B[(kb + r) * 64 + cc];
      asm volatile("global_load_async_to_lds_b128 %0, %1, off"
                   :: "v"(l), "v"(g) : "memory");
    }
  };

  v8f c = {0.f, 0.f, 0.f, 0.f, 0.f, 0.f, 0.f, 0.f};

  stage(0, 0);
  asm volatile("s_wait_asynccnt 0x0" ::: "memory");
  __syncthreads();

  int buf = 0;
  for (int kb = 0; kb < K; kb += 64) {
    if (kb + 64 < K) stage(buf ^ 1, kb + 64);   // prefetch next chunk (async)

    // A operand: lane holds (M=lr, K=k4+2*hl+{0,1})  -> contiguous pair
    // B operand: lane holds (K=k4+2*hl+{0,1}, N=n0+lr)
    #pragma unroll
    for (int k4 = 0; k4 < 64; k4 += 4) {
      v2f a = *(const v2f*)&Ast[buf][lr * APAD + k4 + 2 * hl];
      v2f b;
      b.x = Bst[buf][(k4 + 2 * hl + 0) * APAD + n0 + lr];
      b.y = Bst[buf][(k4 + 2 * hl + 1) * APAD + n0 + lr];
      c = __builtin_amdgcn_wmma_f32_16x16x4_f32(
          /*neg_a=*/false, a, /*neg_b=*/false, b,
          /*c_mod=*/(short)0, c, /*reuse_a=*/false, /*reuse_b=*/false);
    }

    asm volatile("s_wait_asynccnt 0x0" ::: "memory");
    __syncthreads();
    buf ^= 1;
  }

  // C/D layout: c[r] = C[M = r + 8*hl, N = n0 + lr]
  #pragma unroll
  for (int r = 0; r < 8; ++r) {
    int off = (m0 + r + 8 * hl) * 64 + n0 + lr;
    float v = c[r];
    if (mode == 0) {
      out[off] = v;
    } else {
      float var = expf(v) + 1e-6f;
      out[off]  = var;            // feeds next layer (std^2)
      out2[off] = sqrtf(var);     // std output
    }
  }
}

// ---------------------------------------------------------------------------
// Launch
// ---------------------------------------------------------------------------
extern "C" void kernel_launch(void* const* d_in, const int* in_sizes, int n_in,
                              void* d_out, int out_size, void* d_ws, size_t ws_size,
                              hipStream_t stream) {
  (void)in_sizes; (void)n_in; (void)out_size; (void)ws_size;
  const float* x    = (const float*)d_in[0];   // [1024,64]
  const float* adj  = (const float*)d_in[1];   // [1024,1024]
  const float* Wmu  = (const float*)d_in[2];   // [8,64,64]
  const float* Wrho = (const float*)d_in[3];
  const float* eps  = (const float*)d_in[4];
  // d_in[5]/d_in[6] (src/dst) encode the dense graph structure; not needed.

  float* out      = (float*)d_out;
  float* mean_out = out;                 // [1024,64]
  float* std_out  = out + NN * FF;       // [1024,64]
  float* kl_out   = out + 2 * NN * FF;   // scalar

  float* ws  = (float*)d_ws;
  float* W   = ws;                       // 32768
  float* Bm  = W  + WELT;                // 1024*1024
  float* B2  = Bm + NN * NN;             // 1024*1024
  float* T   = B2 + NN * NN;             // 1024*64
  float* mA  = T  + NN * FF;
  float* mB  = mA + NN * FF;
  float* var = mB + NN * FF;
  float* klp = var + NN * FF;            // 128 partials

  // weights + KL
  prep_kernel<<<128, 256, 0, stream>>>(Wmu, Wrho, eps, W, klp);
  kl_final_kernel<<<1, 32, 0, stream>>>(klp, kl_out);

  // adj^T (diag=0) and its square
  transpose_kernel<<<dim3(32, 32), dim3(32, 8), 0, stream>>>(adj, Bm, B2);

  // ---- init layer ----
  gemm_wmma_kernel<<<64, 128, 0, stream>>>(x, FF, W + 0 * 4096, T, nullptr, FF, 0);
  gemm_wmma_kernel<<<64, 128, 0, stream>>>(Bm, NN, T, mA, nullptr, NN, 0);
  gemm_wmma_kernel<<<64, 128, 0, stream>>>(x, FF, W + 1 * 4096, T, nullptr, FF, 0);
  gemm_wmma_kernel<<<64, 128, 0, stream>>>(B2, NN, T, var, std_out, NN, 1);

  // ---- propagation layers ----
  float* hcur = mA;
  float* hnxt = mB;
  for (int i = 0; i < 3; ++i) {
    const float* Wm = W + (2 + 2 * i) * 4096;
    const float* Wv = W + (3 + 2 * i) * 4096;
    float* mdst = (i == 2) ? mean_out : hnxt;
    gemm_wmma_kernel<<<64, 128, 0, stream>>>(hcur, FF, Wm, T, nullptr, FF, 0);
    gemm_wmma_kernel<<<64, 128, 0, stream>>>(Bm, NN, T, mdst, nullptr, NN, 0);
    gemm_wmma_kernel<<<64, 128, 0, stream>>>(var, FF, Wv, T, nullptr, FF, 0);
    gemm_wmma_kernel<<<64, 128, 0, stream>>>(B2, NN, T, var, std_out, NN, 1);
    float* tmp = hcur; hcur = hnxt; hnxt = tmp;
  }
}